// LinearRewardModel_38654705664106
// MI455X (gfx1250) — compile-verified
//
#include <hip/hip_runtime.h>
#include <math.h>
#include <stdint.h>

// Problem constants (match reference): B=8, S=2048, D=4096.
#define Bn 8
#define Sn 2048
#define Dn 4096
#define DSPLIT 4                 // split D across blocks for wave-parallelism
#define DRANGE (Dn / DSPLIT)     // 1024 d-values per block
#define CHUNK 64                 // d-chunk staged per LDS tile
#define NCHUNK (DRANGE / CHUNK)  // 16 iterations
#define ROWS 16                  // s-rows per wave == WMMA M
#define WAVES 8                  // waves per workgroup (wave32)
#define WGROWS (ROWS * WAVES)    // 128 s-rows per workgroup
#define XSTRIDE (CHUNK + 4)      // 68: pad -> conflict-free A-fragment ds reads
#define TILEF (WAVES * ROWS * XSTRIDE)        // floats per tile buffer (8704)
#define LDSF (2 * DRANGE + 2 * TILEF)         // total dynamic LDS floats
#define LDSBYTES (LDSF * 4)                   // 77,824 B (<320KB/WGP)

typedef __attribute__((ext_vector_type(2))) float v2f;
typedef __attribute__((ext_vector_type(4))) float v4f;
typedef __attribute__((ext_vector_type(8))) float v8f;

__device__ __forceinline__ v8f wmma4_f32(v2f a, v2f b, v8f c) {
#if __has_builtin(__builtin_amdgcn_wmma_f32_16x16x4_f32)
  // 8 args: (neg_a, A, neg_b, B, c_mod, C, reuse_a, reuse_b)
  return __builtin_amdgcn_wmma_f32_16x16x4_f32(false, a, false, b, (short)0, c,
                                               false, false);
#else
  asm("v_wmma_f32_16x16x4_f32 %0, %1, %2, %0" : "+v"(c) : "v"(a), "v"(b));
  return c;
#endif
}

// Async DMA: 16 bytes/lane, global -> LDS, tracked by ASYNCcnt (no VGPR data).
__device__ __forceinline__ void async_copy16(const float* gp, const float* lp) {
#if __has_builtin(__builtin_amdgcn_global_load_async_to_lds_b128)
  // Builtin signature: (v4i addrspace(1)*, v4i addrspace(3)*, imm off, imm cpol)
  typedef int v4i_t __attribute__((vector_size(16)));
  typedef __attribute__((address_space(1))) v4i_t* gptr_t;
  typedef __attribute__((address_space(3))) v4i_t* lptr_t;
  // LDS offset is the low 32 bits of the flat pointer (aperture in high bits).
  __builtin_amdgcn_global_load_async_to_lds_b128(
      (gptr_t)(size_t)gp, (lptr_t)(uint32_t)(size_t)lp, 0, 0);
#else
  uint32_t lo = (uint32_t)(size_t)lp;
  uint64_t ga = (uint64_t)(size_t)gp;
  asm volatile("global_load_async_to_lds_b128 %0, %1, off" ::"v"(lo), "v"(ga)
               : "memory");
#endif
}

template <int N>
__device__ __forceinline__ void wait_async() {
#if __has_builtin(__builtin_amdgcn_s_wait_asynccnt)
  __builtin_amdgcn_s_wait_asynccnt(N);
#else
  asm volatile("s_wait_asynccnt %0" ::"i"(N) : "memory");
#endif
  asm volatile("" ::: "memory");  // keep LDS reads below the wait
}

// Kernel A: partial fused[b][s][o] = sum_{d in split} x[b][s][d] * W[o][d].
// Streams x once via double-buffered GLOBAL_LOAD_ASYNC_TO_LDS_B128, then
// accumulates with v_wmma_f32_16x16x4_f32 (W rows in B columns 0..1).
// Each wave owns its 16-row tile: no workgroup barriers in the main loop,
// only per-wave s_wait_asynccnt (async loads retire in order).
__global__ __launch_bounds__(256) void fused_gemv_wmma(
    const float* __restrict__ x, const float* __restrict__ W,
    float* __restrict__ ws) {
  extern __shared__ float lds[];  // [2*DRANGE W slice][tile0][tile1]
  float* ldsW = lds;
  float* tile0 = lds + 2 * DRANGE;

  const int tid  = threadIdx.x;
  const int wave = tid >> 5;
  const int lane = tid & 31;
  const int hi   = lane >> 4;  // half-wave: selects K pair (A and B fragments)
  const int l16  = lane & 15;  // row (A) / column (B)

  const int b     = blockIdx.y;
  const int split = blockIdx.z;
  const int dBase = split * DRANGE;
  const int srow0 = blockIdx.x * WGROWS + wave * ROWS;

  // Stage W[0..1][dBase .. dBase+DRANGE) into LDS once (heavily reused).
  for (int i = tid; i < (2 * DRANGE) / 4; i += 256) {
    const int f  = i * 4;
    const int o  = f / DRANGE;
    const int dl = f % DRANGE;
    *(v4f*)&ldsW[o * DRANGE + dl] =
        *(const v4f*)&W[(size_t)o * Dn + dBase + dl];
  }

  float* buf0 = tile0 + wave * ROWS * XSTRIDE;   // this wave's tile, buffer 0
  float* buf1 = buf0 + TILEF;                    // this wave's tile, buffer 1
  float* bufs[2] = {buf0, buf1};
  const float* xrow = &x[((size_t)b * Sn + srow0 + hi) * Dn + l16 * 4];

  // Issue one 16x64 tile (8 async b128 per lane; lanes cover 2x256B rows).
  auto issue_tile = [&](int ch, float* buf) {
    const int d0 = dBase + ch * CHUNK;
#pragma unroll
    for (int rr = 0; rr < 8; ++rr) {
      async_copy16(xrow + (size_t)(rr * 2) * Dn + d0,
                   &buf[(rr * 2 + hi) * XSTRIDE + l16 * 4]);
    }
  };

  issue_tile(0, bufs[0]);
  __syncthreads();  // ldsW visible to all waves

  v8f c = {};
  const float msk = (l16 < 2) ? 1.0f : 0.0f;  // only B columns 0,1 are live
  for (int ch = 0; ch < NCHUNK; ++ch) {
    float* cur = bufs[ch & 1];
    if (ch + 1 < NCHUNK) {
      issue_tile(ch + 1, bufs[(ch + 1) & 1]);  // prefetch next tile
      wait_async<8>();  // oldest 8 (current tile) complete, 8 in flight
    } else {
      wait_async<0>();
    }
    // A fragment (16x4 f32): lane l16 = row, half-wave selects K={0,1}/{2,3}.
    // B fragment (4x16 f32): lane l16 = column n; W row n for n<2 else 0.
    const float* aBase = &cur[l16 * XSTRIDE + hi * 2];
    const float* bBase = &ldsW[(l16 & 1) * DRANGE + ch * CHUNK + hi * 2];
#pragma unroll
    for (int j = 0; j < CHUNK / 4; ++j) {
      v2f a  = *(const v2f*)&aBase[j * 4];
      v2f bb = (*(const v2f*)&bBase[j * 4]) * msk;  // branchless column mask
      c = wmma4_f32(a, bb, c);
    }
  }

  // C layout: VGPR v holds M=v (lanes 0-15, N=lane) and M=v+8 (lanes 16-31,
  // N=lane-16). Columns N=0,1 -> lanes 0,1,16,17 carry all 16 rows.
  union { v8f v; float f[8]; } u;
  u.v = c;
  if (l16 < 2) {
    const int o     = l16;
    const int mbase = hi * 8;
#pragma unroll
    for (int m = 0; m < 8; ++m) {
      ws[(((size_t)split * Bn + b) * Sn + (srow0 + mbase + m)) * 2 + o] = u.f[m];
    }
  }
}

// Kernel B: combine DSPLIT partials, add bias, sigmoid-gate, mask by length,
// reduce over S per batch. Tiny (1 MB of ws traffic, L2-resident).
__global__ __launch_bounds__(256) void reduce_gate(
    const float* __restrict__ ws, const int* __restrict__ lengths,
    const float* __restrict__ bias, float* __restrict__ out) {
  const int b   = blockIdx.x;
  const int tid = threadIdx.x;
  const int len = lengths[b];
  const float b0 = bias[0], b1 = bias[1];

  float num = 0.0f, den = 0.0f;
  for (int s = tid; s < Sn; s += 256) {
    float f0 = b0, f1 = b1;
#pragma unroll
    for (int sp = 0; sp < DSPLIT; ++sp) {
      const size_t i = (((size_t)sp * Bn + b) * Sn + s) * 2;
      f0 += ws[i];
      f1 += ws[i + 1];
    }
    if (s < len) {
      const float g = 1.0f / (1.0f + __expf(-f0));
      num += g * f1;
      den += g;
    }
  }

  __shared__ float sn[256], sd[256];
  sn[tid] = num;
  sd[tid] = den;
  __syncthreads();
  for (int off = 128; off > 0; off >>= 1) {
    if (tid < off) {
      sn[tid] += sn[tid + off];
      sd[tid] += sd[tid + off];
    }
    __syncthreads();
  }
  if (tid == 0) out[b] = sn[0] / fmaxf(sd[0], 1e-8f);
}

extern "C" void kernel_launch(void* const* d_in, const int* in_sizes, int n_in,
                              void* d_out, int out_size, void* d_ws,
                              size_t ws_size, hipStream_t stream) {
  const float* x       = (const float*)d_in[0];  // (B,S,D) f32
  const int*   lengths = (const int*)d_in[1];    // (B,) i32
  const float* W       = (const float*)d_in[2];  // (2,D) f32
  const float* bias    = (const float*)d_in[3];  // (2,) f32
  float* ws  = (float*)d_ws;                     // needs 512 KiB
  float* out = (float*)d_out;                    // (B,) f32

  fused_gemv_wmma<<<dim3(Sn / WGROWS, Bn, DSPLIT), 256, LDSBYTES, stream>>>(
      x, W, ws);
  reduce_gate<<<Bn, 256, 0, stream>>>(ws, lengths, bias, out);
}